// MambaClassifier_7937099563533
// MI455X (gfx1250) — compile-verified
//
#include <hip/hip_runtime.h>
#include <hip/hip_bf16.h>

#define B_ 32
#define L_ 1024
#define DIN_ 16
#define DM_ 512
#define DI_ 1024
#define H_ 16
#define P_ 64
#define N_ 64
#define CONV_DIM_ 1152
#define NXB_ 1168            // CONV_DIM + H   (xBC || dt), all we need for every token
#define DPROJ_ 2192
#define M_ (B_ * L_)         // 32768 tokens
#define NCLS_ 10
#define EPS_ 1e-5f

typedef float v2f __attribute__((ext_vector_type(2)));
typedef float v8f __attribute__((ext_vector_type(8)));

__device__ __forceinline__ float silu_f(float x) { return x / (1.0f + expf(-x)); }

// ---------------------------------------------------------------------------
// fp32 GEMM on matrix cores: C[M x N] = act(A[M x K] @ W[K x ldw][colOff..) + bias)
// Each wave computes a 16x64 output slab (4 accumulators) via
// V_WMMA_F32_16X16X4_F32, reusing one A-fragment across 4 WMMAs per k-step:
//   - A traffic / 4 (the dominant L2 stream for the big projection GEMM)
//   - 9 VMEM loads per 4 wmma instead of 3 per 1.
// A-fragment (ISA 16x4 fp32): lanes 0-15 -> K=k0,k0+1 ; lanes 16-31 -> K=k0+2,k0+3.
// B-fragment: elem0 = row k0 (lanes<16) / k0+2 (lanes>=16); elem1 = k0+1 / k0+3.
// C/D: VGPR i -> row m0+i (lanes<16) or m0+8+i (lanes>=16), col n0+(lane&15).
// N-tile groups with <4 tiles (only the N=1168 edge) take a scalar-tile fallback.
// ---------------------------------------------------------------------------
__global__ void __launch_bounds__(128)
gemm_wmma_f32(const float* __restrict__ A, const float* __restrict__ W,
              const float* __restrict__ bias, float* __restrict__ C,
              int M, int K, int N, int ldw, int colOff, int doRelu)
{
    const int wave = threadIdx.x >> 5;
    const int lane = threadIdx.x & 31;
    const int tilesN  = N >> 4;
    const int groups  = (tilesN + 3) >> 2;          // 4 N-tiles per wave
    const int job = blockIdx.x * 4 + wave;
    if (job >= (M >> 4) * groups) return;           // wave-uniform exit

    const int mt = job / groups;
    const int ng = job % groups;
    const int m0 = mt << 4;
    const int n0 = ng << 6;                          // base column of the 64-wide slab
    const int cnt = min(4, tilesN - (ng << 2));      // active 16-wide tiles in slab
    const int lo = lane & 15;
    const int hi = lane >> 4;

    const float* __restrict__ arow  = A + (size_t)(m0 + lo) * K + 2 * hi;
    const float* __restrict__ wbase = W + (size_t)(2 * hi) * ldw + colOff + n0 + lo;

    if (cnt == 4) {
        v8f acc0 = {0.f,0.f,0.f,0.f,0.f,0.f,0.f,0.f};
        v8f acc1 = acc0, acc2 = acc0, acc3 = acc0;
        for (int k0 = 0; k0 < K; k0 += 4) {
            v2f a; a.x = arow[0]; a.y = arow[1];
            v2f b0; b0.x = wbase[0];  b0.y = wbase[ldw];
            v2f b1; b1.x = wbase[16]; b1.y = wbase[ldw + 16];
            v2f b2; b2.x = wbase[32]; b2.y = wbase[ldw + 32];
            v2f b3; b3.x = wbase[48]; b3.y = wbase[ldw + 48];
            acc0 = __builtin_amdgcn_wmma_f32_16x16x4_f32(false, a, false, b0, (short)0, acc0, false, false);
            acc1 = __builtin_amdgcn_wmma_f32_16x16x4_f32(false, a, false, b1, (short)0, acc1, false, false);
            acc2 = __builtin_amdgcn_wmma_f32_16x16x4_f32(false, a, false, b2, (short)0, acc2, false, false);
            acc3 = __builtin_amdgcn_wmma_f32_16x16x4_f32(false, a, false, b3, (short)0, acc3, false, false);
            arow += 4;
            wbase += (size_t)4 * ldw;
        }
        float* __restrict__ crow = C + (size_t)(m0 + 8 * hi) * N + n0 + lo;
        v8f accs[4] = {acc0, acc1, acc2, acc3};
#pragma unroll
        for (int t = 0; t < 4; ++t) {
            const float bv = bias ? bias[n0 + t * 16 + lo] : 0.0f;
#pragma unroll
            for (int i = 0; i < 8; ++i) {
                float v = accs[t][i] + bv;
                if (doRelu) v = fmaxf(v, 0.0f);
                crow[(size_t)i * N + t * 16] = v;
            }
        }
    } else {
        // Edge slab (only the last group of the N=1168 GEMM): one tile at a time.
        for (int t = 0; t < cnt; ++t) {
            const float* ar = arow;
            const float* wc = wbase + t * 16;
            v8f acc = {0.f,0.f,0.f,0.f,0.f,0.f,0.f,0.f};
            for (int k0 = 0; k0 < K; k0 += 4) {
                v2f a; a.x = ar[0]; a.y = ar[1];
                v2f b; b.x = wc[0]; b.y = wc[ldw];
                acc = __builtin_amdgcn_wmma_f32_16x16x4_f32(false, a, false, b, (short)0, acc, false, false);
                ar += 4;
                wc += (size_t)4 * ldw;
            }
            const float bv = bias ? bias[n0 + t * 16 + lo] : 0.0f;
            float* __restrict__ crow = C + (size_t)(m0 + 8 * hi) * N + n0 + t * 16 + lo;
#pragma unroll
            for (int i = 0; i < 8; ++i) {
                float v = acc[i] + bv;
                if (doRelu) v = fmaxf(v, 0.0f);
                crow[(size_t)i * N] = v;
            }
        }
    }
}

// ---------------------------------------------------------------------------
// Sequential selective scan, one block (64 threads) per (b, h).
// Fuses the depthwise causal conv (k=4) + SiLU on the fly (raw projection is L2
// resident), keeps state S[p][0..63] in VGPRs, broadcasts the convolved B-row
// through LDS. Only the last timestep's y is emitted (that's all the head uses).
// ---------------------------------------------------------------------------
__global__ void __launch_bounds__(64)
scan_kernel(const float* __restrict__ raw,      // M x NXB_ : [xBC_raw | dt_raw]
            const float* __restrict__ conv_w,   // 1152 x 4
            const float* __restrict__ conv_b,   // 1152
            const float* __restrict__ dt_bias,  // 16
            const float* __restrict__ A_log,    // 16
            const float* __restrict__ Dvec,     // 16
            float* __restrict__ y_last)         // B x 1024
{
    const int b = blockIdx.x >> 4;
    const int h = blockIdx.x & 15;
    const int p = threadIdx.x;               // 0..63
    const float* __restrict__ base = raw + (size_t)b * L_ * NXB_;

    const int cx = h * P_ + p;               // this thread's x channel
    const float wx0 = conv_w[cx * 4 + 0], wx1 = conv_w[cx * 4 + 1];
    const float wx2 = conv_w[cx * 4 + 2], wx3 = conv_w[cx * 4 + 3];
    const float bx = conv_b[cx];

    const int cB = DI_ + p;                  // B channel this thread convolves
    const float wb0 = conv_w[cB * 4 + 0], wb1 = conv_w[cB * 4 + 1];
    const float wb2 = conv_w[cB * 4 + 2], wb3 = conv_w[cB * 4 + 3];
    const float bB = conv_b[cB];

    const float dtb  = dt_bias[h];
    const float negA = -expf(A_log[h]);

    float S[64];
#pragma unroll
    for (int n = 0; n < 64; ++n) S[n] = 0.0f;

    __shared__ float sB[64];
    __shared__ float sC[64];

    float xv = 0.0f;
    for (int l = 0; l < L_; ++l) {
        const float* r0 = base + (size_t)l * NXB_;
        // causal conv for own x channel: out = sum_k raw[l-3+k]*w[k]
        xv = bx + r0[cx] * wx3;
        if (l >= 1) xv += r0[cx - NXB_] * wx2;
        if (l >= 2) xv += r0[cx - 2 * NXB_] * wx1;
        if (l >= 3) xv += r0[cx - 3 * NXB_] * wx0;
        xv = silu_f(xv);
        // conv for B channel p
        float bvv = bB + r0[cB] * wb3;
        if (l >= 1) bvv += r0[cB - NXB_] * wb2;
        if (l >= 2) bvv += r0[cB - 2 * NXB_] * wb1;
        if (l >= 3) bvv += r0[cB - 3 * NXB_] * wb0;
        bvv = silu_f(bvv);
        // dt -> softplus -> dA
        const float dtr = r0[CONV_DIM_ + h] + dtb;
        const float dt = (dtr > 20.0f) ? dtr : log1pf(expf(dtr));
        const float dA = expf(negA * dt);

        __syncthreads();                      // previous iter done reading sB
        sB[p] = bvv;
        __syncthreads();

        const float coef = dt * xv;
#pragma unroll
        for (int n = 0; n < 64; ++n) S[n] = fmaf(S[n], dA, coef * sB[n]);
    }

    // y only needed at l = L-1: convolve C there and contract.
    const int cC = DI_ + N_ + p;
    const float* rl = base + (size_t)(L_ - 1) * NXB_;
    float cv = conv_b[cC]
             + rl[cC]            * conv_w[cC * 4 + 3]
             + rl[cC - NXB_]     * conv_w[cC * 4 + 2]
             + rl[cC - 2 * NXB_] * conv_w[cC * 4 + 1]
             + rl[cC - 3 * NXB_] * conv_w[cC * 4 + 0];
    cv = silu_f(cv);
    __syncthreads();
    sC[p] = cv;
    __syncthreads();

    float y = 0.0f;
#pragma unroll
    for (int n = 0; n < 64; ++n) y = fmaf(S[n], sC[n], y);
    y += Dvec[h] * xv;                        // + D * xs at last step
    y_last[(size_t)b * DI_ + cx] = y;
}

// ---------------------------------------------------------------------------
// Last-token head: z = u_last @ Win[:, :DI], gate, RMS-norm, @Wout, @Wc + bc.
// One block per batch element.
// ---------------------------------------------------------------------------
__global__ void __launch_bounds__(256)
finalize_kernel(const float* __restrict__ u,      // M x DM
                const float* __restrict__ Win,    // DM x DPROJ
                const float* __restrict__ y_last, // B x DI
                const float* __restrict__ norm_w, // DI
                const float* __restrict__ Wout,   // DI x DM
                const float* __restrict__ Wc,     // DM x NCLS
                const float* __restrict__ bc,     // NCLS
                float* __restrict__ out)          // B x NCLS
{
    const int b = blockIdx.x;
    const int t = threadIdx.x;
    __shared__ float su[DM_];
    __shared__ float syn[DI_];
    __shared__ float sseq[DM_];
    __shared__ float rbuf[256];

    const float* urow = u + ((size_t)b * L_ + (L_ - 1)) * DM_;
    for (int i = t; i < DM_; i += 256) su[i] = urow[i];
    __syncthreads();

    float ss = 0.0f;
    for (int j = t; j < DI_; j += 256) {
        float z = 0.0f;
        for (int k = 0; k < DM_; ++k) z = fmaf(su[k], Win[(size_t)k * DPROJ_ + j], z);
        const float yg = y_last[(size_t)b * DI_ + j] * silu_f(z);
        syn[j] = yg;
        ss = fmaf(yg, yg, ss);
    }
    rbuf[t] = ss;
    __syncthreads();
    for (int s = 128; s > 0; s >>= 1) {
        if (t < s) rbuf[t] += rbuf[t + s];
        __syncthreads();
    }
    const float scale = rsqrtf(rbuf[0] / (float)DI_ + EPS_);

    for (int j = t; j < DI_; j += 256) syn[j] *= scale * norm_w[j];
    __syncthreads();

    for (int d = t; d < DM_; d += 256) {
        float s = 0.0f;
        for (int j = 0; j < DI_; ++j) s = fmaf(syn[j], Wout[(size_t)j * DM_ + d], s);
        sseq[d] = s;
    }
    __syncthreads();

    if (t < NCLS_) {
        float s = bc[t];
        for (int k = 0; k < DM_; ++k) s = fmaf(sseq[k], Wc[(size_t)k * NCLS_ + t], s);
        out[(size_t)b * NCLS_ + t] = s;
    }
}

// ---------------------------------------------------------------------------
extern "C" void kernel_launch(void* const* d_in, const int* in_sizes, int n_in,
                              void* d_out, int out_size, void* d_ws, size_t ws_size,
                              hipStream_t stream)
{
    const float* x       = (const float*)d_in[0];
    const float* W1      = (const float*)d_in[1];
    const float* b1      = (const float*)d_in[2];
    const float* W2      = (const float*)d_in[3];
    const float* b2      = (const float*)d_in[4];
    const float* W3      = (const float*)d_in[5];
    const float* b3      = (const float*)d_in[6];
    const float* Win     = (const float*)d_in[7];
    const float* conv_w  = (const float*)d_in[8];
    const float* conv_b  = (const float*)d_in[9];
    const float* dt_bias = (const float*)d_in[10];
    const float* A_log   = (const float*)d_in[11];
    const float* Dv      = (const float*)d_in[12];
    const float* norm_w  = (const float*)d_in[13];
    const float* Wout    = (const float*)d_in[14];
    const float* Wc      = (const float*)d_in[15];
    const float* bc      = (const float*)d_in[16];

    float* ws    = (float*)d_ws;
    float* h1    = ws;                          // M x 128
    float* h2    = h1 + (size_t)M_ * 128;       // M x 256
    float* uu    = h2 + (size_t)M_ * 256;       // M x 512
    float* raw   = uu + (size_t)M_ * DM_;       // M x 1168 (xBC || dt)
    float* ylast = raw + (size_t)M_ * NXB_;     // B x 1024

    auto nblk = [](int M, int N) {
        int groups = ((N >> 4) + 3) >> 2;
        int jobs = (M >> 4) * groups;
        return (jobs + 3) / 4;
    };

    // MLP stack
    gemm_wmma_f32<<<nblk(M_, 128), 128, 0, stream>>>(x,  W1, b1, h1, M_, DIN_, 128, 128, 0, 1);
    gemm_wmma_f32<<<nblk(M_, 256), 128, 0, stream>>>(h1, W2, b2, h2, M_, 128, 256, 256, 0, 1);
    gemm_wmma_f32<<<nblk(M_, DM_), 128, 0, stream>>>(h2, W3, b3, uu, M_, 256, DM_, DM_, 0, 0);
    // Projection: only xBC||dt columns (skip z for all but last token — saves ~47% of the big GEMM)
    gemm_wmma_f32<<<nblk(M_, NXB_), 128, 0, stream>>>(uu, Win, nullptr, raw, M_, DM_, NXB_, DPROJ_, DI_, 0);
    // Sequential scan with fused conv+SiLU; emits y only at the last timestep
    scan_kernel<<<B_ * H_, 64, 0, stream>>>(raw, conv_w, conv_b, dt_bias, A_log, Dv, ylast);
    // Last-token head (z matvec + gate + RMS-norm + Wout + classifier)
    finalize_kernel<<<B_, 256, 0, stream>>>(uu, Win, ylast, norm_w, Wout, Wc, bc, (float*)d_out);
}